// Mamba2ForMapping_31705448579636
// MI455X (gfx1250) — compile-verified
//
#include <hip/hip_runtime.h>
#include <hip/hip_bf16.h>

#define B_ 8
#define S_ 8192
#define D_ 768
#define V_ 64
#define L_ 26

#define DBLK   128            // columns per D-block
#define NDBLK  (D_ / DBLK)    // 6
#define CHUNKS 8              // token chunks per batch
#define TOKS_PER_CHUNK (S_ / CHUNKS)   // 1024

typedef float f4  __attribute__((ext_vector_type(4)));
typedef float v2f __attribute__((ext_vector_type(2)));
typedef float v8f __attribute__((ext_vector_type(8)));

// ---------------------------------------------------------------------------
// Kernel 1: per-(batch, d-block, chunk) segment sum via LDS ds_add_f32 atomics,
// flushed with global_atomic_add_f32. Streams hidden_states exactly once.
// ---------------------------------------------------------------------------
__global__ void __launch_bounds__(256) seg_reduce_kernel(
    const float* __restrict__ hidden,   // [B, S, D]
    const int*   __restrict__ ids,      // [B, S]
    float*       __restrict__ sums,     // [B*V, D]
    float*       __restrict__ counts)   // [B*V]
{
    __shared__ float acc[V_ * DBLK];    // 32 KB
    __shared__ float cnt[V_];

    const int tid  = threadIdx.x;
    const int lane = tid & 31;
    const int wave = tid >> 5;          // 0..7

    int blk = blockIdx.x;
    const int chunk = blk % CHUNKS;  blk /= CHUNKS;
    const int dblk  = blk % NDBLK;   blk /= NDBLK;
    const int batch = blk;

    for (int i = tid; i < V_ * DBLK; i += 256) acc[i] = 0.0f;
    if (tid < V_) cnt[tid] = 0.0f;
    __syncthreads();

    const size_t rowBase = (size_t)batch * S_;
    const int    col     = dblk * DBLK + lane * 4;
    const int    tok0    = chunk * TOKS_PER_CHUNK;

    for (int t = tok0 + wave; t < tok0 + TOKS_PER_CHUNK; t += 8) {
        const int v = ids[rowBase + t];
        const f4  h = __builtin_nontemporal_load(
            (const f4*)(hidden + ((rowBase + t) * (size_t)D_ + col)));
        float* row = &acc[v * DBLK + lane * 4];
        atomicAdd(&row[0], h.x);        // ds_add_f32 (no return)
        atomicAdd(&row[1], h.y);
        atomicAdd(&row[2], h.z);
        atomicAdd(&row[3], h.w);
        if (dblk == 0 && lane == 0) atomicAdd(&cnt[v], 1.0f);
    }
    __syncthreads();

    float* sdst = sums + ((size_t)batch * V_) * D_ + dblk * DBLK;
    for (int i = tid; i < V_ * DBLK; i += 256) {
        const int v = i / DBLK, c = i % DBLK;
        const float val = acc[i];
        if (val != 0.0f) atomicAdd(&sdst[(size_t)v * D_ + c], val);
    }
    if (dblk == 0 && tid < V_) {
        const float cv = cnt[tid];
        if (cv != 0.0f) atomicAdd(&counts[batch * V_ + tid], cv);
    }
}

// ---------------------------------------------------------------------------
// Kernel 2: means = sums / max(counts,1); logits = means @ W.T + b via
// V_WMMA_F32_16X16X4_F32 (fp32 matrix pipe), masked store to [B, V, L].
// M = 512, N = 26 -> 2 tiles of 16, K = 768 -> 192 steps of 4.
// A 16x4 f32 layout: lanes 0-15 M=lane hold K=0,1; lanes 16-31 hold K=2,3.
// B 4x16 f32 layout: lanes 0-15 N=lane hold K=0,1; lanes 16-31 hold K=2,3.
// C/D 16x16:  vgpr r = row r (lanes 0-15) / row r+8 (lanes 16-31), N = lane&15.
// ---------------------------------------------------------------------------
__global__ void __launch_bounds__(256) wmma_classify_kernel(
    const float* __restrict__ sums,     // [B*V, D]
    const float* __restrict__ counts,   // [B*V]
    const float* __restrict__ W,        // [L, D]
    const float* __restrict__ bias,     // [L]
    float*       __restrict__ out)      // [B*V, L]
{
    const int tid  = threadIdx.x;
    const int wave = tid >> 5;
    const int lane = tid & 31;
    const int half = lane >> 4;         // 0: K+0/1, 1: K+2/3
    const int l15  = lane & 15;

    const int tile  = blockIdx.x * 8 + wave;   // 0..63
    const int mtile = tile >> 1;               // 0..31
    const int ntile = tile & 1;                // 0..1

    const int   rowA = mtile * 16 + l15;       // mean row this lane feeds A with
    const int   n    = ntile * 16 + l15;       // class column this lane feeds B with
    const float inv  = 1.0f / fmaxf(counts[rowA], 1.0f);

    const float* arow = sums + (size_t)rowA * D_;
    const float* brow = W + (size_t)(n < L_ ? n : 0) * D_;
    const float  bsel = (n < L_) ? 1.0f : 0.0f;   // zero-pad classes 26..31
    const int    koff = half * 2;

    v8f c = {0.f, 0.f, 0.f, 0.f, 0.f, 0.f, 0.f, 0.f};
    #pragma unroll 4
    for (int k = 0; k < D_; k += 4) {
        v2f a, b2;
        a.x  = arow[k + koff + 0] * inv;
        a.y  = arow[k + koff + 1] * inv;
        b2.x = brow[k + koff + 0] * bsel;
        b2.y = brow[k + koff + 1] * bsel;
        c = __builtin_amdgcn_wmma_f32_16x16x4_f32(
            /*neg_a=*/false, a, /*neg_b=*/false, b2,
            /*c_mod=*/(short)0, c, /*reuse_a=*/false, /*reuse_b=*/false);
    }

    if (n < L_) {
        const float bn = bias[n];
        #pragma unroll
        for (int r = 0; r < 8; ++r) {
            const int  m       = mtile * 16 + r + half * 8;    // output row
            const bool present = (counts[m] > 0.0f) && ((m & (V_ - 1)) != 0);
            out[(size_t)m * L_ + n] = present ? (c[r] + bn) : 0.0f;
        }
    }
}

// ---------------------------------------------------------------------------
extern "C" void kernel_launch(void* const* d_in, const int* in_sizes, int n_in,
                              void* d_out, int out_size, void* d_ws, size_t ws_size,
                              hipStream_t stream) {
    const float* hidden = (const float*)d_in[0];   // [B,S,D] f32
    const float* W      = (const float*)d_in[1];   // [L,D]   f32
    const float* bias   = (const float*)d_in[2];   // [L]     f32
    const int*   ids    = (const int*)d_in[3];     // [B,S]   i32
    float*       out    = (float*)d_out;           // [B,V,L] f32

    float* sums   = (float*)d_ws;                        // B*V*D
    float* counts = sums + (size_t)B_ * V_ * D_;         // B*V
    const size_t zbytes = ((size_t)B_ * V_ * D_ + (size_t)B_ * V_) * sizeof(float);
    hipMemsetAsync(d_ws, 0, zbytes, stream);

    seg_reduce_kernel<<<dim3(B_ * NDBLK * CHUNKS), 256, 0, stream>>>(
        hidden, ids, sums, counts);
    wmma_classify_kernel<<<dim3(8), 256, 0, stream>>>(
        sums, counts, W, bias, out);
}